// GAEModel_53730040873121
// MI455X (gfx1250) — compile-verified
//
#include <hip/hip_runtime.h>

// GCN (2-layer) for MI455X / gfx1250, wave32.
// Phase split: fp32 WMMA GEMM (W staged in LDS, pair-interleaved by K so each
// B fragment is one ds_load_b64), L2-resident edge scatter-add via
// global_atomic_add_f32, elementwise finalize (self-loop + bias + relu).

typedef __attribute__((ext_vector_type(2))) float v2f;
typedef __attribute__((ext_vector_type(8))) float v8f;

#define D 128  // D_IN == D_HID == 128

// ---------------- degree / normalization ----------------

__global__ void init_deg_kernel(float* __restrict__ deg, int n) {
  int i = blockIdx.x * blockDim.x + threadIdx.x;
  if (i < n) deg[i] = 1.0f;  // self-loop contribution
}

__global__ void deg_edges_kernel(const int* __restrict__ dst, float* __restrict__ deg, int e) {
  int i = blockIdx.x * blockDim.x + threadIdx.x;
  if (i < e) atomicAdd(&deg[dst[i]], 1.0f);
}

__global__ void rsqrt_kernel(float* __restrict__ deg, int n) {
  int i = blockIdx.x * blockDim.x + threadIdx.x;
  if (i < n) deg[i] = rsqrtf(deg[i]);  // deg >= 1 always
}

__global__ void zero_kernel(float* __restrict__ p, size_t n) {
  size_t i = blockIdx.x * (size_t)blockDim.x + threadIdx.x;
  if (i < n) p[i] = 0.0f;
}

// ---------------- GEMM: C[M,128] = A[M,128] * W[128,128] via WMMA f32 16x16x4 ----------------
// One wave computes a 16x128 strip of C (8 N-tiles). A fragment: one b64 global
// load per k-step. W is staged in LDS pair-interleaved by K:
//   Wl[(k>>1)*256 + 2*c + (k&1)] = W[k][c]
// so the B fragment (W[k][c], W[k+1][c]) is a single aligned ds_load_b64 that
// lands directly in an even VGPR pair for WMMA (no register shuffling).
__global__ __launch_bounds__(128) void gemm128_wmma_kernel(const float* __restrict__ A,
                                                           const float* __restrict__ W,
                                                           float* __restrict__ C,
                                                           int mtiles) {
  __shared__ float Wl[D * D];  // 64 KB, pair-interleaved layout
  {
    const float4* Wg4 = (const float4*)W;
    for (int i = threadIdx.x; i < (D * D) / 4; i += blockDim.x) {
      float4 w = Wg4[i];
      int k = (i << 2) >> 7;   // source row (K index)
      int c = (i << 2) & 127;  // source column
      float* dstb = &Wl[(k >> 1) * (2 * D) + (k & 1)];
      dstb[2 * (c + 0)] = w.x;
      dstb[2 * (c + 1)] = w.y;
      dstb[2 * (c + 2)] = w.z;
      dstb[2 * (c + 3)] = w.w;
    }
  }
  __syncthreads();

  int wave = (int)((blockIdx.x * blockDim.x + threadIdx.x) >> 5);
  int lane = threadIdx.x & 31;
  if (wave >= mtiles) return;  // whole-wave exit: EXEC stays all-ones for WMMA

  int m  = lane & 15;   // row within A tile / column within B tile
  int kh = lane >> 4;   // K-half select per ISA 16x4 fp32 A/B layout

  const float* arow = A + (size_t)(wave * 16 + m) * D;

  v8f acc[8] = {};
  for (int k0 = 0; k0 < D; k0 += 4) {
    int k = k0 + 2 * kh;          // lanes 0-15: K=k0,k0+1 ; lanes 16-31: K=k0+2,k0+3
    int p = k >> 1;               // k-pair row in LDS
    v2f a = *(const v2f*)(arow + k);
    const float* brow = &Wl[p * (2 * D) + 2 * m];
#pragma unroll
    for (int nt = 0; nt < 8; ++nt) {
      v2f b = *(const v2f*)(brow + 32 * nt);  // (W[k][col], W[k+1][col]) in one b64
      acc[nt] = __builtin_amdgcn_wmma_f32_16x16x4_f32(
          /*neg_a=*/false, a, /*neg_b=*/false, b,
          /*c_mod=*/(short)0, acc[nt], /*reuse_a=*/false, /*reuse_b=*/false);
    }
  }

  // C/D layout: VGPR i -> rows i (lanes 0-15) and i+8 (lanes 16-31), col = lane%16
#pragma unroll
  for (int nt = 0; nt < 8; ++nt) {
#pragma unroll
    for (int i = 0; i < 8; ++i) {
      C[(size_t)(wave * 16 + i + 8 * kh) * D + (nt * 16 + m)] = acc[nt][i];
    }
  }
}

// ---------------- edge aggregation: out[dst] += dinv[s]*dinv[d] * h[src] ----------------
// One wave per edge; each lane carries a float4 (32 lanes x 4 = 128 channels).
__global__ void agg_edges_kernel(const float* __restrict__ h, const int* __restrict__ src,
                                 const int* __restrict__ dst, const float* __restrict__ dinv,
                                 float* __restrict__ out, int e) {
  int w = (int)((blockIdx.x * (size_t)blockDim.x + threadIdx.x) >> 5);
  int lane = threadIdx.x & 31;
  if (w >= e) return;
  int s = src[w];
  int d = dst[w];
  float norm = dinv[s] * dinv[d];
  float4 v = *(const float4*)(h + (size_t)s * D + lane * 4);
  float* o = out + (size_t)d * D + lane * 4;
  atomicAdd(o + 0, norm * v.x);
  atomicAdd(o + 1, norm * v.y);
  atomicAdd(o + 2, norm * v.z);
  atomicAdd(o + 3, norm * v.w);
}

// ---------------- finalize: agg += dinv^2 * h_lin (self-loop) + bias; optional relu ----------------
__global__ void finalize_kernel(float* __restrict__ agg, const float* __restrict__ hlin,
                                const float* __restrict__ dinv, const float* __restrict__ bias,
                                int n, int do_relu) {
  size_t idx = blockIdx.x * (size_t)blockDim.x + threadIdx.x;
  if (idx >= (size_t)n * D) return;
  int node = (int)(idx >> 7);
  int c = (int)(idx & (D - 1));
  float di = dinv[node];
  float v = agg[idx] + di * di * hlin[idx] + bias[c];
  if (do_relu) v = fmaxf(v, 0.0f);
  agg[idx] = v;
}

// ---------------- host-side launch ----------------

extern "C" void kernel_launch(void* const* d_in, const int* in_sizes, int n_in,
                              void* d_out, int out_size, void* d_ws, size_t ws_size,
                              hipStream_t stream) {
  const float* x  = (const float*)d_in[0];
  const int*   ei = (const int*)d_in[1];   // [2, E] int32 row-major
  const float* W1 = (const float*)d_in[2];
  const float* b1 = (const float*)d_in[3];
  const float* W2 = (const float*)d_in[4];
  const float* b2 = (const float*)d_in[5];

  const int n = in_sizes[0] / D;   // 50000 (multiple of 16)
  const int e = in_sizes[1] / 2;   // 800000
  const int* src = ei;
  const int* dst = ei + e;
  float* out = (float*)d_out;

  // workspace carve-out (floats): dinv | H1 | AGG1 | H2  (~77 MB total)
  float* ws = (float*)d_ws;
  const size_t ndpad = ((size_t)n + 255) & ~(size_t)255;
  const size_t nd = (size_t)n * D;
  float* dinv = ws;
  float* H1   = ws + ndpad;
  float* AGG1 = H1 + nd;
  float* H2   = AGG1 + nd;

  const int mtiles = n / 16;  // n is a multiple of 16 for this workload

  const int thr = 256;
  const int nBlocksN  = (n + thr - 1) / thr;
  const int nBlocksE  = (e + thr - 1) / thr;
  const int nBlocksND = (int)((nd + thr - 1) / thr);
  const int nBlocksEW = (e + 7) / 8;            // 8 waves (edges) per 256-thr block
  const int nBlocksG  = (mtiles + 3) / 4;       // 4 waves (M-tiles) per 128-thr block

  // symmetric normalization (shared by both layers)
  init_deg_kernel<<<nBlocksN, thr, 0, stream>>>(dinv, n);
  deg_edges_kernel<<<nBlocksE, thr, 0, stream>>>(dst, dinv, e);
  rsqrt_kernel<<<nBlocksN, thr, 0, stream>>>(dinv, n);

  // layer 1: H1 = x@W1 ; AGG1 = scatter(norm*H1) ; AGG1 = relu(AGG1 + dinv^2*H1 + b1)
  gemm128_wmma_kernel<<<nBlocksG, 128, 0, stream>>>(x, W1, H1, mtiles);
  zero_kernel<<<nBlocksND, thr, 0, stream>>>(AGG1, nd);
  agg_edges_kernel<<<nBlocksEW, thr, 0, stream>>>(H1, src, dst, dinv, AGG1, e);
  finalize_kernel<<<nBlocksND, thr, 0, stream>>>(AGG1, H1, dinv, b1, n, /*relu=*/1);

  // layer 2: H2 = AGG1@W2 ; out = scatter(norm*H2) ; out = out + dinv^2*H2 + b2
  gemm128_wmma_kernel<<<nBlocksG, 128, 0, stream>>>(AGG1, W2, H2, mtiles);
  zero_kernel<<<nBlocksND, thr, 0, stream>>>(out, nd);
  agg_edges_kernel<<<nBlocksEW, thr, 0, stream>>>(H2, src, dst, dinv, out, e);
  finalize_kernel<<<nBlocksND, thr, 0, stream>>>(out, H2, dinv, b2, n, /*relu=*/0);
}